// ODEFunction_70849780514973
// MI455X (gfx1250) — compile-verified
//
#include <hip/hip_runtime.h>

#ifndef __has_builtin
#define __has_builtin(x) 0
#endif

// ---------------- tunables ----------------
#define TILE    512   // edges staged per LDS tile (src+dst int64 = 8KB/buffer)
#define THREADS 256   // 8 waves (wave32) per block
// D_FEAT = 64 floats/row = 16 float4 chunks; 16 threads cooperate per edge.

// ---------------- CDNA5 async-to-LDS path (guarded) ----------------
#if __has_builtin(__builtin_amdgcn_global_load_async_to_lds_b128)
#define HAVE_ASYNC_COPY 1
#else
#define HAVE_ASYNC_COPY 0
#endif

#if __has_builtin(__builtin_amdgcn_s_wait_asynccnt)
#define WAIT_ASYNC(n) __builtin_amdgcn_s_wait_asynccnt(n)
#elif defined(__AMDGCN__)
#define WAIT_ASYNC(n) asm volatile("s_wait_asynccnt %0" ::"i"(n) : "memory")
#else
#define WAIT_ASYNC(n) ((void)0)
#endif

// int32x4 pointers in explicit address spaces for the async builtin.
typedef int v4i __attribute__((vector_size(4 * sizeof(int))));
typedef __attribute__((address_space(1))) v4i* gv4i_p;  // global (AS1)
typedef __attribute__((address_space(3))) v4i* lv4i_p;  // LDS (AS3)

__device__ __forceinline__ void atom_add_f32(float* p, float v) {
  // Lowers to non-returning global_atomic_add_f32 (L2 RMW), not a CAS loop.
  unsafeAtomicAdd(p, v);
}

// ---------------- kernel 0: zero the output ----------------
__global__ void lg_zero_kernel(float4* __restrict__ out, int n4) {
  int i = blockIdx.x * blockDim.x + threadIdx.x;
  if (i < n4) out[i] = make_float4(0.f, 0.f, 0.f, 0.f);
}

// ---------------- index tile staging ----------------
// Stages TILE edges' src (int64) and dst (int64) into LDS.
// Every thread issues exactly 2 async b128 loads (uniform -> exact ASYNCcnt
// bookkeeping); tail addresses are clamped to stay inside edge_index[2*E].
__device__ __forceinline__ void stage_tile(unsigned long long* shBuf,
                                           const char* eiBytes,
                                           unsigned long long maxOff,
                                           long long e0, int E, int t) {
  unsigned long long off0 =
      (unsigned long long)e0 * 8ull + (unsigned long long)t * 16ull;            // src region
  unsigned long long off1 =
      ((unsigned long long)E + (unsigned long long)e0) * 8ull +
      (unsigned long long)t * 16ull;                                            // dst region
  if (off0 > maxOff) off0 = maxOff;
  if (off1 > maxOff) off1 = maxOff;
#if HAVE_ASYNC_COPY
  __builtin_amdgcn_global_load_async_to_lds_b128(
      (gv4i_p)(eiBytes + off0), (lv4i_p)(shBuf + 2 * t),
      /*offset=*/0, /*cpol=*/0);
  __builtin_amdgcn_global_load_async_to_lds_b128(
      (gv4i_p)(eiBytes + off1), (lv4i_p)(shBuf + TILE + 2 * t),
      /*offset=*/0, /*cpol=*/0);
#else
  const unsigned long long* g0 = (const unsigned long long*)(eiBytes + off0);
  const unsigned long long* g1 = (const unsigned long long*)(eiBytes + off1);
  shBuf[2 * t]            = g0[0];
  shBuf[2 * t + 1]        = g0[1];
  shBuf[TILE + 2 * t]     = g1[0];
  shBuf[TILE + 2 * t + 1] = g1[1];
#endif
}

// ---------------- kernel 1: gather + atomic scatter ----------------
// 16 threads per edge, float4 per thread (256B row = 2 cachelines, coalesced).
// x (25.6MB) and out (25.6MB) stay resident in the 192MB L2; atomics resolve at L2.
__global__ void __launch_bounds__(THREADS)
lgconv_scatter_kernel(const float4* __restrict__ x4,
                      const long long* __restrict__ ei,
                      float* __restrict__ out, int E) {
  __shared__ unsigned long long sh[2][2 * TILE];  // 16KB: [buf][src qwords | dst qwords]

  const int t     = threadIdx.x;
  const int laneE = t >> 4;   // which of 16 edges in a row-pass
  const int chunk = t & 15;   // which float4 of the 64-float row

  const char* eiBytes = (const char*)ei;
  const unsigned long long maxOff =
      (unsigned long long)2 * (unsigned long long)E * 8ull - 16ull;
  const int ntiles = (E + TILE - 1) / TILE;

  int tile = blockIdx.x;
  if (tile >= ntiles) return;

  // prologue: stage first tile
  stage_tile(sh[0], eiBytes, maxOff, (long long)tile * TILE, E, t);

  int buf = 0;
  for (; tile < ntiles; tile += gridDim.x, buf ^= 1) {
    const int nextTile = tile + (int)gridDim.x;
    if (nextTile < ntiles) {
      // double-buffer: issue next tile's 2 async loads, then wait until only
      // those 2 remain outstanding -> current buffer complete (in-order).
      stage_tile(sh[buf ^ 1], eiBytes, maxOff, (long long)nextTile * TILE, E, t);
      WAIT_ASYNC(2);
    } else {
      WAIT_ASYNC(0);
    }
    __syncthreads();  // make all waves' staged LDS data visible

    const unsigned* sh32 = (const unsigned*)sh[buf];
    const long long e0   = (long long)tile * TILE;
#pragma unroll 4
    for (int it = 0; it < TILE / 16; ++it) {
      const int el      = it * 16 + laneE;
      const long long e = e0 + el;
      if (e >= E) break;                         // tail guard (el monotonic)
      const unsigned s = sh32[2 * el];           // low dword of int64 src
      const unsigned d = sh32[2 * (TILE + el)];  // low dword of int64 dst
      const float4 v = x4[(size_t)s * 16 + chunk];           // global_load_b128
      float* op = out + (size_t)d * 64 + (size_t)chunk * 4;
      atom_add_f32(op + 0, v.x);                 // global_atomic_add_f32 x4
      atom_add_f32(op + 1, v.y);
      atom_add_f32(op + 2, v.z);
      atom_add_f32(op + 3, v.w);
    }
    __syncthreads();  // all reads of sh[buf] done before it is re-staged
  }
}

// ---------------- host entry ----------------
extern "C" void kernel_launch(void* const* d_in, const int* in_sizes, int n_in,
                              void* d_out, int out_size, void* d_ws, size_t ws_size,
                              hipStream_t stream) {
  // d_in[0] = t (scalar, unused), d_in[1] = x [N,64] f32, d_in[2] = edge_index [2,E] i64
  const float* x = (const float*)d_in[1];
  const long long* ei = (const long long*)d_in[2];
  float* out = (float*)d_out;
  const int E = in_sizes[2] / 2;

  const int n4 = out_size / 4;  // out_size = N*64, divisible by 4
  lg_zero_kernel<<<(n4 + THREADS - 1) / THREADS, THREADS, 0, stream>>>(
      (float4*)out, n4);

  const int ntiles = (E + TILE - 1) / TILE;
  const int blocks = ntiles < 2048 ? ntiles : 2048;
  lgconv_scatter_kernel<<<blocks, THREADS, 0, stream>>>(
      (const float4*)x, ei, out, E);
}